// ResidualVectorQuantize_1022202217163
// MI455X (gfx1250) — compile-verified
//
#include <hip/hip_runtime.h>
#include <hip/hip_bf16.h>
#include <math.h>

// Problem constants (match reference)
#define B_    16
#define D_IN  512
#define T_    4096
#define N_CB  9
#define K_CB  1024
#define D_CB  8
#define EPSF  1e-12f

#define TCOLS 16            // time-columns per block
#define NGRP  16            // channel groups per column
#define CH_PT 32            // channels per thread (512 / 16)

typedef __attribute__((ext_vector_type(2))) float v2f;
typedef __attribute__((ext_vector_type(8))) float v8f;
typedef __attribute__((ext_vector_type(4))) unsigned int v4u;
typedef __attribute__((ext_vector_type(8))) int v8i;
typedef __attribute__((ext_vector_type(4))) int v4i;

union V8U { v8f v; float f[8]; };

#if __has_builtin(__builtin_amdgcn_wmma_f32_16x16x4_f32)
#define HAS_WMMA_F32_4 1
#endif
#if __has_builtin(__builtin_amdgcn_tensor_load_to_lds)
#define HAS_TDM 1
#endif

#ifdef HAS_TDM
// ---------------------------------------------------------------------------
// Tensor Data Mover: 1-D (single-row 2-D) tile load, global -> LDS.
// D# per CDNA5 ISA ch.8:
//  group0: [1:0]=count=1, [63:32]=lds_addr, [120:64]=global_addr, [127:126]=type=2
//  group1: [17:16]=data_size(2 -> 4B), [79:48]=tensor_dim0, [111:80]=tensor_dim1,
//          [127:112]=tile_dim0, [143:128]=tile_dim1(0=unused),
//          [207:160]=tensor_dim0_stride
//  remaining groups: zero (tensor <= 2D)
// nelem: number of f32 elements (must fit tile_dim0's 16 bits).
// This toolchain (clang-23 / therock-10.0) exposes the 6-arg builtin:
//   (uint32x4, int32x8, int32x4, int32x4, int32x8, i32 cpol)
// ---------------------------------------------------------------------------
__device__ __forceinline__ void tdm_load_1d(unsigned int lds_addr,
                                            const void* gptr,
                                            unsigned int nelem)
{
    const unsigned long long ga = (unsigned long long)(uintptr_t)gptr;
    v4u g0;
    g0.x = 1u;                                                   // count=1 (valid)
    g0.y = lds_addr;                                             // LDS byte address
    g0.z = (unsigned int)ga;                                     // global_addr[31:0]
    g0.w = (unsigned int)((ga >> 32) & 0x01FFFFFFu) | (2u << 30);// addr[56:32] | type=2
    v8i g1;
    g1[0] = (int)(2u << 16);                                     // data_size = 4 bytes
    g1[1] = (int)((nelem & 0xFFFFu) << 16);                      // tensor_dim0[15:0]
    g1[2] = (int)(((nelem >> 16) & 0xFFFFu) | (1u << 16));       // tensor_dim0[31:16] | tensor_dim1=1
    g1[3] = (int)((nelem & 0xFFFFu) << 16);                      // tile_dim0 = nelem
    g1[4] = 0;                                                   // tile_dim1/2 = 0 (unused)
    g1[5] = (int)nelem;                                          // tensor_dim0_stride[31:0]
    g1[6] = 0;
    g1[7] = 0;
    const v4i z4 = {0, 0, 0, 0};
    const v8i z8 = {0, 0, 0, 0, 0, 0, 0, 0};
    __builtin_amdgcn_tensor_load_to_lds(g0, g1, z4, z4, z8, 0);
}
#endif

// ---------------------------------------------------------------------------
// Prep kernel 1: weight-norm for in_proj, stored TRANSPOSED as WinT[s][c][k]
// so the main kernel reads one contiguous float8 per channel.
// ---------------------------------------------------------------------------
__global__ __launch_bounds__(256)
void rvq_prep_win(const float* __restrict__ in_v, const float* __restrict__ in_g,
                  float* __restrict__ WinT)
{
    __shared__ float red[256];
    const int row = blockIdx.x;            // s*8 + k
    const int tid = threadIdx.x;
    const float* vrow = in_v + (size_t)row * D_IN;
    float sum = 0.f;
    for (int c = tid; c < D_IN; c += 256) { float v = vrow[c]; sum += v * v; }
    red[tid] = sum;
    __syncthreads();
    for (int off = 128; off >= 1; off >>= 1) {
        if (tid < off) red[tid] += red[tid + off];
        __syncthreads();
    }
    const float scale = in_g[row] / sqrtf(red[0]);
    const int s = row >> 3, k = row & 7;
    float* dst = WinT + (size_t)s * D_IN * D_CB;
    for (int c = tid; c < D_IN; c += 256)
        dst[(size_t)c * D_CB + k] = vrow[c] * scale;
}

// ---------------------------------------------------------------------------
// Prep kernel 2: weight-norm for out_proj rows, normalized codebook cbN +
// per-entry ||cb_n||^2, and zero the loss accumulators.
// ---------------------------------------------------------------------------
__global__ __launch_bounds__(256)
void rvq_prep_rows(const float* __restrict__ out_v, const float* __restrict__ out_g,
                   const float* __restrict__ codebooks,
                   float* __restrict__ WoutN, float* __restrict__ cbN,
                   float* __restrict__ c2, float* __restrict__ out)
{
    const int gt = blockIdx.x * blockDim.x + threadIdx.x;
    if (gt == 0) {
        float* losses = out + (size_t)B_ * N_CB * T_ + (size_t)B_ * D_IN * T_;
        losses[0] = 0.f;
        losses[1] = 0.f;
    }
    if (gt < N_CB * D_IN) {
        const float* v = out_v + (size_t)gt * D_CB;
        float tmp[D_CB]; float n2 = 0.f;
        #pragma unroll
        for (int k = 0; k < D_CB; ++k) { tmp[k] = v[k]; n2 += tmp[k] * tmp[k]; }
        const float scale = out_g[gt] / sqrtf(n2);   // reference: direct division
        #pragma unroll
        for (int k = 0; k < D_CB; ++k) WoutN[(size_t)gt * D_CB + k] = tmp[k] * scale;
    } else if (gt < N_CB * D_IN + N_CB * K_CB) {
        const int r = gt - N_CB * D_IN;              // s*1024 + n
        const float* cb = codebooks + (size_t)r * D_CB;
        float tmp[D_CB]; float n2 = 0.f;
        #pragma unroll
        for (int k = 0; k < D_CB; ++k) { tmp[k] = cb[k]; n2 += tmp[k] * tmp[k]; }
        const float inv = 1.f / fmaxf(sqrtf(n2), EPSF);
        float s2 = 0.f;
        #pragma unroll
        for (int k = 0; k < D_CB; ++k) {
            float vn = tmp[k] * inv;
            cbN[(size_t)r * D_CB + k] = vn;
            s2 += vn * vn;
        }
        c2[r] = s2;
    }
}

// ---------------------------------------------------------------------------
// Fused main kernel: one block = 16 time columns of one batch. Residual and
// latent live in VGPRs across all 9 codebook steps (z read once from HBM,
// latent written once). Per step, the TDM stages WinT[s], WoutN[s], c2[s]
// into LDS (one DMA each, TENSORcnt-tracked) so the 16-way per-column reuse
// is served from LDS instead of redundant global loads. Distance GEMM uses
// V_WMMA_F32_16X16X4_F32; argmin(dist) == argmax(2*s - ||cb_n||^2).
// ---------------------------------------------------------------------------
__global__ __launch_bounds__(256)
void rvq_fused(const float* __restrict__ z,
               const float* __restrict__ in_b,
               const float* __restrict__ out_b,
               const float* __restrict__ codebooks,
               const float* __restrict__ WinT,
               const float* __restrict__ WoutN,
               const float* __restrict__ cbN,
               const float* __restrict__ c2,
               float* __restrict__ out)
{
    const int tid  = threadIdx.x;
    const int col  = tid & 15;       // time column within tile (coalesced t)
    const int g    = tid >> 4;       // channel group (32 channels each)
    const int lane = tid & 31;
    const int wave = tid >> 5;       // 8 waves per block
    const int b    = blockIdx.y;
    const int t0   = blockIdx.x * TCOLS;

    __shared__ float s_win[D_IN * D_CB];         // 16 KB staged in-proj weights
    __shared__ float s_wout[D_IN * D_CB];        // 16 KB staged out-proj weights
    __shared__ float s_c2[K_CB];                 //  4 KB staged ||cb_n||^2
    __shared__ float s_pe[TCOLS * NGRP * D_CB];  //  8 KB partial in-proj sums
    __shared__ float s_ze[TCOLS * D_CB];         // z_e (pre-norm)
    __shared__ float s_en[TCOLS * D_CB];         // normalized encodings
    __shared__ float s_zq[TCOLS * D_CB];         // gathered raw codebook vecs
    __shared__ float s_wb[8 * 16];               // per-wave best scores
    __shared__ int   s_wi[8 * 16];               // per-wave best indices
    __shared__ int   s_idx[TCOLS];
    __shared__ float s_loss[TCOLS * D_CB];

    float res[CH_PT];
    float lat[CH_PT];

    const size_t zbase = (size_t)b * D_IN * T_ + (size_t)(t0 + col);
    #pragma unroll
    for (int j = 0; j < CH_PT; ++j) {
        const int ch = g * CH_PT + j;
        res[j] = z[zbase + (size_t)ch * T_];
        lat[j] = 0.f;
    }

    float* latent = out + (size_t)B_ * N_CB * T_;
    float* losses = latent + (size_t)B_ * D_IN * T_;

    for (int s = 0; s < N_CB; ++s) {
        // ---- phase 0: TDM-stage this step's weights into LDS ----
#ifdef HAS_TDM
        if (wave == 0) {   // wave-uniform branch: only wave 0 issues the DMAs
            tdm_load_1d((unsigned int)(uintptr_t)(void*)s_win,
                        WinT + (size_t)s * D_IN * D_CB, D_IN * D_CB);
            tdm_load_1d((unsigned int)(uintptr_t)(void*)s_wout,
                        WoutN + (size_t)s * D_IN * D_CB, D_IN * D_CB);
            tdm_load_1d((unsigned int)(uintptr_t)(void*)s_c2,
                        c2 + (size_t)s * K_CB, K_CB);
            __builtin_amdgcn_s_wait_tensorcnt(0);
        }
#else
        for (int i = tid; i < D_IN * D_CB; i += 256) {
            s_win[i]  = WinT [(size_t)s * D_IN * D_CB + i];
            s_wout[i] = WoutN[(size_t)s * D_IN * D_CB + i];
        }
        for (int i = tid; i < K_CB; i += 256)
            s_c2[i] = c2[(size_t)s * K_CB + i];
#endif
        __syncthreads();

        // ---- phase 1: in_proj partials over this thread's 32 channels ----
        float p[D_CB];
        #pragma unroll
        for (int k = 0; k < D_CB; ++k) p[k] = 0.f;
        #pragma unroll 4
        for (int j = 0; j < CH_PT; ++j) {
            const int ch = g * CH_PT + j;
            const float4* wp = (const float4*)(s_win + (size_t)ch * D_CB);
            const float4 w0 = wp[0], w1 = wp[1];
            const float r = res[j];
            p[0] = fmaf(w0.x, r, p[0]); p[1] = fmaf(w0.y, r, p[1]);
            p[2] = fmaf(w0.z, r, p[2]); p[3] = fmaf(w0.w, r, p[3]);
            p[4] = fmaf(w1.x, r, p[4]); p[5] = fmaf(w1.y, r, p[5]);
            p[6] = fmaf(w1.z, r, p[6]); p[7] = fmaf(w1.w, r, p[7]);
        }
        #pragma unroll
        for (int k = 0; k < D_CB; ++k)
            s_pe[(col * NGRP + g) * D_CB + k] = p[k];
        __syncthreads();

        // ---- phase 2a: reduce groups -> z_e ----
        if (tid < TCOLS * D_CB) {
            const int c = tid >> 3, k = tid & 7;
            float acc = 0.f;
            #pragma unroll
            for (int gg = 0; gg < NGRP; ++gg)
                acc += s_pe[(c * NGRP + gg) * D_CB + k];
            acc += in_b[s * D_CB + k];
            s_ze[c * D_CB + k] = acc;
        }
        __syncthreads();

        // ---- phase 2b: L2-normalize each length-8 encoding ----
        if (tid < TCOLS) {
            float n2 = 0.f;
            #pragma unroll
            for (int k = 0; k < D_CB; ++k) { float v = s_ze[tid * D_CB + k]; n2 += v * v; }
            const float inv = 1.f / fmaxf(sqrtf(n2), EPSF);
            #pragma unroll
            for (int k = 0; k < D_CB; ++k)
                s_en[tid * D_CB + k] = s_ze[tid * D_CB + k] * inv;
        }
        __syncthreads();

        // ---- phase 3: WMMA distance GEMM + argmax(2*s - c2) ----
        // A (16x4 f32): lanes 0-15 hold K={0,1}, lanes 16-31 hold K={2,3}; M = lane&15
        const int m16 = lane & 15;
        const int kh  = (lane >> 4) * 2;
        v2f a0, a1;
        a0[0] = s_en[m16 * D_CB + kh + 0]; a0[1] = s_en[m16 * D_CB + kh + 1];
        a1[0] = s_en[m16 * D_CB + kh + 4]; a1[1] = s_en[m16 * D_CB + kh + 5];

        float best[8]; int bidx[8];
        #pragma unroll
        for (int r = 0; r < 8; ++r) { best[r] = -3.0e38f; bidx[r] = 0; }

        const float* cbn_s = cbN + (size_t)s * K_CB * D_CB;

        for (int nt = wave * 8; nt < wave * 8 + 8; ++nt) {   // 64 N-tiles / 8 waves
            const int n = nt * 16 + m16;
            const float* cp = cbn_s + (size_t)n * D_CB;
            v2f b0, b1;
            b0[0] = cp[kh + 0]; b0[1] = cp[kh + 1];
            b1[0] = cp[kh + 4]; b1[1] = cp[kh + 5];
            V8U acc;
            #pragma unroll
            for (int r = 0; r < 8; ++r) acc.f[r] = 0.f;
#ifdef HAS_WMMA_F32_4
            acc.v = __builtin_amdgcn_wmma_f32_16x16x4_f32(
                false, a0, false, b0, (short)0, acc.v, false, false);
            acc.v = __builtin_amdgcn_wmma_f32_16x16x4_f32(
                false, a1, false, b1, (short)0, acc.v, false, false);
#else
            #pragma unroll
            for (int r = 0; r < 8; ++r) {
                const int mm = r + ((lane >> 4) << 3);
                float sd = 0.f;
                #pragma unroll
                for (int k = 0; k < D_CB; ++k)
                    sd = fmaf(s_en[mm * D_CB + k], cp[k], sd);
                acc.f[r] = sd;
            }
#endif
            const float c2v = s_c2[n];
            #pragma unroll
            for (int r = 0; r < 8; ++r) {
                const float sc = 2.f * acc.f[r] - c2v;
                if (sc > best[r] || (sc == best[r] && n < bidx[r])) {
                    best[r] = sc; bidx[r] = n;
                }
            }
        }
        // C layout: lanes 0-15 -> M = r, lanes 16-31 -> M = r+8; reduce over N
        // across the 16 lanes of each half (butterfly, width 16).
        #pragma unroll
        for (int off = 8; off >= 1; off >>= 1) {
            #pragma unroll
            for (int r = 0; r < 8; ++r) {
                const float os = __shfl_xor(best[r], off, 16);
                const int   oi = __shfl_xor(bidx[r], off, 16);
                if (os > best[r] || (os == best[r] && oi < bidx[r])) {
                    best[r] = os; bidx[r] = oi;
                }
            }
        }
        if (m16 == 0) {
            const int mb = (lane >> 4) * 8;
            #pragma unroll
            for (int r = 0; r < 8; ++r) {
                s_wb[wave * 16 + mb + r] = best[r];
                s_wi[wave * 16 + mb + r] = bidx[r];
            }
        }
        __syncthreads();
        if (tid < TCOLS) {
            float bb = s_wb[tid]; int bi = s_wi[tid];
            #pragma unroll
            for (int w = 1; w < 8; ++w) {
                const float os = s_wb[w * 16 + tid];
                const int   oi = s_wi[w * 16 + tid];
                if (os > bb || (os == bb && oi < bi)) { bb = os; bi = oi; }
            }
            s_idx[tid] = bi;
            // codes[b][s][t] as float
            out[(size_t)b * N_CB * T_ + (size_t)s * T_ + (t0 + tid)] = (float)bi;
        }
        __syncthreads();

        // ---- phase 4: gather raw z_q, per-step commit/codebook loss ----
        if (tid < TCOLS * D_CB) {
            const int c = tid >> 3, k = tid & 7;
            const float zq = codebooks[((size_t)s * K_CB + s_idx[c]) * D_CB + k];
            s_zq[c * D_CB + k] = zq;
            const float d = s_ze[c * D_CB + k] - zq;
            s_loss[tid] = d * d;
        }
        __syncthreads();
        if (tid == 0) {
            float acc = 0.f;
            for (int i = 0; i < TCOLS * D_CB; ++i) acc += s_loss[i];
            const float contrib = acc * (1.0f / (float)(B_ * D_CB * T_));
            atomicAdd(&losses[0], contrib);   // closs
            atomicAdd(&losses[1], contrib);   // bloss (numerically identical)
        }

        // prefetch next step's normalized codebook (global_prefetch_b8)
        if (s + 1 < N_CB && tid < 64) {
            __builtin_prefetch(cbN + (size_t)(s + 1) * K_CB * D_CB + tid * 128, 0, 1);
        }

        // ---- phase 5: out_proj + residual/latent update (rank-8, VALU) ----
        float zqc[D_CB];
        #pragma unroll
        for (int k = 0; k < D_CB; ++k) zqc[k] = s_zq[col * D_CB + k];
        #pragma unroll 4
        for (int j = 0; j < CH_PT; ++j) {
            const int ch = g * CH_PT + j;
            const float4* wp = (const float4*)(s_wout + (size_t)ch * D_CB);
            const float4 w0 = wp[0], w1 = wp[1];
            float acc = out_b[s * D_IN + ch];
            acc = fmaf(w0.x, zqc[0], acc); acc = fmaf(w0.y, zqc[1], acc);
            acc = fmaf(w0.z, zqc[2], acc); acc = fmaf(w0.w, zqc[3], acc);
            acc = fmaf(w1.x, zqc[4], acc); acc = fmaf(w1.y, zqc[5], acc);
            acc = fmaf(w1.z, zqc[6], acc); acc = fmaf(w1.w, zqc[7], acc);
            lat[j] += acc;
            res[j] -= acc;
        }
        __syncthreads();   // protect shared buffers before next step's TDM
    }

    // ---- write latent once ----
    #pragma unroll
    for (int j = 0; j < CH_PT; ++j) {
        const int ch = g * CH_PT + j;
        latent[zbase + (size_t)ch * T_] = lat[j];
    }
}

// ---------------------------------------------------------------------------
extern "C" void kernel_launch(void* const* d_in, const int* in_sizes, int n_in,
                              void* d_out, int out_size, void* d_ws, size_t ws_size,
                              hipStream_t stream)
{
    const float* z         = (const float*)d_in[0];
    const float* in_v      = (const float*)d_in[1];
    const float* in_g      = (const float*)d_in[2];
    const float* in_b      = (const float*)d_in[3];
    const float* out_v     = (const float*)d_in[4];
    const float* out_g     = (const float*)d_in[5];
    const float* out_b     = (const float*)d_in[6];
    const float* codebooks = (const float*)d_in[7];
    float* out = (float*)d_out;

    // workspace layout (floats): WinT | WoutN | cbN | c2  (~627 KB total)
    float* ws    = (float*)d_ws;
    float* WinT  = ws;                                   // 9*512*8
    float* WoutN = WinT + (size_t)N_CB * D_IN * D_CB;    // 9*512*8
    float* cbN   = WoutN + (size_t)N_CB * D_IN * D_CB;   // 9*1024*8
    float* c2    = cbN + (size_t)N_CB * K_CB * D_CB;     // 9*1024

    rvq_prep_win<<<N_CB * D_CB, 256, 0, stream>>>(in_v, in_g, WinT);

    const int rows = N_CB * D_IN + N_CB * K_CB;
    rvq_prep_rows<<<(rows + 255) / 256, 256, 0, stream>>>(out_v, out_g, codebooks,
                                                          WoutN, cbN, c2, out);

    dim3 grid(T_ / TCOLS, B_);   // 256 x 16 = 4096 blocks, 8 waves each
    rvq_fused<<<grid, 256, 0, stream>>>(z, in_b, out_b, codebooks,
                                        WinT, WoutN, cbN, c2, out);
}